// MGCN_26645977104437
// MI455X (gfx1250) — compile-verified
//
#include <hip/hip_runtime.h>
#include <math.h>

#define Nn   4096
#define FEAT 200
#define OUTC 200

typedef __attribute__((ext_vector_type(2))) float v2f;
typedef __attribute__((ext_vector_type(8))) float v8f;

#define BM 128
#define BN 128
#define KB 16

// ---------------------------------------------------------------------------
// Generic fp32 WMMA GEMM: C(opt +=) = epilogue(accScale*A@B(+T) + bias + addScale*addM)
// 256 threads = 8 waves; block tile 128x128; wave tile 32x64 (2x4 WMMA 16x16 tiles).
// Fast path (all dims tile-divisible): double-buffered GLOBAL_LOAD_ASYNC_TO_LDS_B128
// staging with s_wait_asynccnt pipelining. Fallback: guarded scalar staging.
// ---------------------------------------------------------------------------
__global__ __launch_bounds__(256) void gemm_wmma_f32(
    const float* __restrict__ A, int lda,
    const float* __restrict__ B, int ldb, int transB,
    float* __restrict__ C, int ldc,
    int Mdim, int Ndim, int Kdim,
    float accScale,
    const float* __restrict__ bias,                 // len Ndim or null
    const float* __restrict__ addM, int ldadd, float addScale,  // or null
    const float* __restrict__ rowScale,             // len Mdim or null (multiplicative)
    const float* __restrict__ colScale,             // len Ndim or null (multiplicative)
    int applySigmoid, int accumulateC)
{
    // Single shared object => LDS offsets of smem start at 0 (needed for async ops).
    __shared__ float smem[2 * BM * KB + 2 * KB * BN];   // 32 KB
    const int SA0 = 0;
    const int SA1 = BM * KB;
    const int SB0 = 2 * BM * KB;
    const int SB1 = 2 * BM * KB + KB * BN;

    const int t    = threadIdx.x;
    const int lane = t & 31;
    const int wid  = t >> 5;
    const int wm0  = (wid & 3) * 32;   // wave M origin in block tile
    const int wn0  = (wid >> 2) * 64;  // wave N origin in block tile
    const int hi   = lane >> 4;        // 0/1 half-wave
    const int l16  = lane & 15;

    const int blockM = blockIdx.y * BM;
    const int blockN = blockIdx.x * BN;

    v8f acc[2][4] = {};

    // ---- consume one staged K-chunk from LDS with WMMA ----
    auto computeChunk = [&](int aBase, int bBase) {
        #pragma unroll
        for (int kk = 0; kk < KB; kk += 4) {
            const int ko = kk + 2 * hi;
            v2f afrag[2];
            #pragma unroll
            for (int tm = 0; tm < 2; ++tm) {
                int row = wm0 + tm * 16 + l16;
                afrag[tm].x = smem[aBase + row * KB + ko];
                afrag[tm].y = smem[aBase + row * KB + ko + 1];
            }
            v2f bfrag[4];
            #pragma unroll
            for (int tn = 0; tn < 4; ++tn) {
                int col = wn0 + tn * 16 + l16;
                if (transB) {            // n-major LDS layout [BN][KB]
                    bfrag[tn].x = smem[bBase + col * KB + ko];
                    bfrag[tn].y = smem[bBase + col * KB + ko + 1];
                } else {                 // k-major LDS layout [KB][BN]
                    bfrag[tn].x = smem[bBase + ko * BN + col];
                    bfrag[tn].y = smem[bBase + (ko + 1) * BN + col];
                }
            }
            #pragma unroll
            for (int tm = 0; tm < 2; ++tm)
                #pragma unroll
                for (int tn = 0; tn < 4; ++tn)
                    acc[tm][tn] = __builtin_amdgcn_wmma_f32_16x16x4_f32(
                        false, afrag[tm], false, bfrag[tn],
                        (short)0, acc[tm][tn], false, false);
        }
    };

    const bool fast = ((Mdim & (BM - 1)) == 0) && ((Ndim & (BN - 1)) == 0) &&
                      ((Kdim & (KB - 1)) == 0);

    if (fast) {
        // ---- async staging: 4 x GLOBAL_LOAD_ASYNC_TO_LDS_B128 per wave per chunk ----
        auto issueChunk = [&](int aBase, int bBase, int k0) {
            #pragma unroll
            for (int j = 0; j < 2; ++j) {          // A tile: 128x16 f32, 16B chunks
                int c   = t + 256 * j;
                int row = c >> 2, kq = c & 3;
                unsigned go = (unsigned)(((size_t)(blockM + row) * lda + k0 + 4 * kq) * 4u);
                unsigned lo = (unsigned)((aBase + row * KB + 4 * kq) * 4);
                asm volatile("global_load_async_to_lds_b128 %0, %1, %2"
                             :: "v"(lo), "v"(go), "s"(A) : "memory");
            }
            #pragma unroll
            for (int j = 0; j < 2; ++j) {          // B tile
                int c = t + 256 * j;
                unsigned go, lo;
                if (transB) {                      // [BN][KB] layout
                    int n = c >> 2, kq = c & 3;
                    go = (unsigned)(((size_t)(blockN + n) * ldb + k0 + 4 * kq) * 4u);
                    lo = (unsigned)((bBase + n * KB + 4 * kq) * 4);
                } else {                           // [KB][BN] layout
                    int kk = c >> 5, nq = c & 31;
                    go = (unsigned)(((size_t)(k0 + kk) * ldb + blockN + 4 * nq) * 4u);
                    lo = (unsigned)((bBase + kk * BN + 4 * nq) * 4);
                }
                asm volatile("global_load_async_to_lds_b128 %0, %1, %2"
                             :: "v"(lo), "v"(go), "s"(B) : "memory");
            }
        };

        const int nch = Kdim / KB;
        issueChunk(SA0, SB0, 0);
        for (int i = 0; i < nch; ++i) {
            const int cur = i & 1;
            if (i + 1 < nch) {
                issueChunk(cur ? SA0 : SA1, cur ? SB0 : SB1, (i + 1) * KB);
                asm volatile("s_wait_asynccnt 0x4" ::: "memory");  // older 4 done
            } else {
                asm volatile("s_wait_asynccnt 0x0" ::: "memory");
            }
            __syncthreads();
            computeChunk(cur ? SA1 : SA0, cur ? SB1 : SB0);
            __syncthreads();   // all waves done reading before buffer reuse
        }
    } else {
        // ---- guarded scalar staging (partial tiles / ragged K) ----
        for (int k0 = 0; k0 < Kdim; k0 += KB) {
            #pragma unroll
            for (int i = 0; i < 8; ++i) {
                int e = t * 8 + i;
                int row = e >> 4, kk = e & 15;
                int gm = blockM + row, gk = k0 + kk;
                float v = 0.f;
                if (gm < Mdim && gk < Kdim) v = A[(size_t)gm * lda + gk];
                smem[SA0 + row * KB + kk] = v;
            }
            #pragma unroll
            for (int i = 0; i < 8; ++i) {
                int e = t * 8 + i;
                if (transB) {                      // [BN][KB] layout
                    int n = e >> 4, kk = e & 15;
                    int gn = blockN + n, gk = k0 + kk;
                    float v = 0.f;
                    if (gk < Kdim && gn < Ndim) v = B[(size_t)gn * ldb + gk];
                    smem[SB0 + n * KB + kk] = v;
                } else {                           // [KB][BN] layout
                    int kk = e >> 7, n = e & 127;
                    int gn = blockN + n, gk = k0 + kk;
                    float v = 0.f;
                    if (gk < Kdim && gn < Ndim) v = B[(size_t)gk * ldb + gn];
                    smem[SB0 + kk * BN + n] = v;
                }
            }
            __syncthreads();
            computeChunk(SA0, SB0);
            __syncthreads();
        }
    }

    // Epilogue per §7.12.2 C/D layout: VGPR r -> M = r + 8*hi, N = l16
    #pragma unroll
    for (int tm = 0; tm < 2; ++tm) {
        #pragma unroll
        for (int tn = 0; tn < 4; ++tn) {
            #pragma unroll
            for (int r = 0; r < 8; ++r) {
                int gm = blockM + wm0 + tm * 16 + r + 8 * hi;
                int gn = blockN + wn0 + tn * 16 + l16;
                if (gm < Mdim && gn < Ndim) {
                    float v = accScale * acc[tm][tn][r];
                    if (bias)     v += bias[gn];
                    if (addM)     v += addScale * addM[(size_t)gm * ldadd + gn];
                    if (applySigmoid) v = 1.0f / (1.0f + __expf(-v));
                    if (rowScale) v *= rowScale[gm];
                    if (colScale) v *= colScale[gn];
                    size_t o = (size_t)gm * ldc + gn;
                    if (accumulateC) C[o] += v; else C[o] = v;
                }
            }
        }
    }
}

// rnrm[i] = 1 / max(||attr_i||, 1e-8)
__global__ __launch_bounds__(256) void row_invnorm(const float* __restrict__ A, int n,
                                                   float* __restrict__ rnrm) {
    __shared__ float red[256];
    int i = blockIdx.x;
    float s = 0.f;
    for (int j = threadIdx.x; j < n; j += 256) {
        float v = A[(size_t)i * n + j];
        s += v * v;
    }
    red[threadIdx.x] = s; __syncthreads();
    for (int w = 128; w > 0; w >>= 1) {
        if (threadIdx.x < w) red[threadIdx.x] += red[threadIdx.x + w];
        __syncthreads();
    }
    if (threadIdx.x == 0) rnrm[i] = 1.0f / fmaxf(sqrtf(red[0]), 1e-8f);
}

// In place: A[i,:] /= max(A[i,:]); deg[i] = sum_{j!=i} A[i,j]; dinv = rsqrt(deg) or 0
__global__ __launch_bounds__(256) void rowmax_norm_deg(float* __restrict__ A, int n,
                                                       float* __restrict__ dinv) {
    __shared__ float red[256];
    int i = blockIdx.x;
    float m = -3.0e38f;
    for (int j = threadIdx.x; j < n; j += 256)
        m = fmaxf(m, A[(size_t)i * n + j]);
    red[threadIdx.x] = m; __syncthreads();
    for (int w = 128; w > 0; w >>= 1) {
        if (threadIdx.x < w) red[threadIdx.x] = fmaxf(red[threadIdx.x], red[threadIdx.x + w]);
        __syncthreads();
    }
    float inv = 1.0f / red[0];
    __syncthreads();
    float s = 0.f;
    for (int j = threadIdx.x; j < n; j += 256) {
        float v = A[(size_t)i * n + j] * inv;
        A[(size_t)i * n + j] = v;
        if (j != i) s += v;
    }
    red[threadIdx.x] = s; __syncthreads();
    for (int w = 128; w > 0; w >>= 1) {
        if (threadIdx.x < w) red[threadIdx.x] += red[threadIdx.x + w];
        __syncthreads();
    }
    if (threadIdx.x == 0) dinv[i] = (red[0] > 0.f) ? rsqrtf(red[0]) : 0.f;
}

// M[i,j] = (i==j) ? 0 : -dinv[i]*dinv[j]*An[j,i]   (tiled transpose read)
__global__ __launch_bounds__(256) void build_M(const float* __restrict__ An,
                                               const float* __restrict__ dinv,
                                               float* __restrict__ Mo, int n) {
    __shared__ float tile[32][33];
    int bx = blockIdx.x * 32, by = blockIdx.y * 32;
    int tx = threadIdx.x & 31, ty = threadIdx.x >> 5;  // ty in 0..7
    for (int r = ty; r < 32; r += 8)
        tile[r][tx] = An[(size_t)(bx + r) * n + (by + tx)];   // tile[p][q]=An[bx+p][by+q]
    __syncthreads();
    for (int r = ty; r < 32; r += 8) {
        int i = by + r, j = bx + tx;
        float v = tile[tx][r];                                // An[j][i]
        Mo[(size_t)i * n + j] = (i == j) ? 0.f : (-dinv[i] * dinv[j] * v);
    }
}

__global__ __launch_bounds__(256) void bn_stats(const float* __restrict__ S, int rows, int cols,
                                                float* __restrict__ mean, float* __restrict__ istd) {
    __shared__ float s1[256], s2[256];
    int c = blockIdx.x;
    float su = 0.f, sq = 0.f;
    for (int r = threadIdx.x; r < rows; r += 256) {
        float v = S[(size_t)r * cols + c];
        su += v; sq += v * v;
    }
    s1[threadIdx.x] = su; s2[threadIdx.x] = sq; __syncthreads();
    for (int w = 128; w > 0; w >>= 1) {
        if (threadIdx.x < w) { s1[threadIdx.x] += s1[threadIdx.x + w]; s2[threadIdx.x] += s2[threadIdx.x + w]; }
        __syncthreads();
    }
    if (threadIdx.x == 0) {
        float m = s1[0] / rows;
        float var = s2[0] / rows - m * m;     // biased variance
        mean[c] = m;
        istd[c] = rsqrtf(var + 1e-5f);
    }
}

__global__ __launch_bounds__(256) void bn_apply(float* __restrict__ S, int rows, int cols,
                                                const float* __restrict__ mean,
                                                const float* __restrict__ istd,
                                                const float* __restrict__ gamma,
                                                const float* __restrict__ beta) {
    size_t idx = (size_t)blockIdx.x * 256 + threadIdx.x;
    size_t total = (size_t)rows * cols;
    if (idx < total) {
        int c = (int)(idx % cols);
        S[idx] = gamma[c] * (S[idx] - mean[c]) * istd[c] + beta[c];
    }
}

extern "C" void kernel_launch(void* const* d_in, const int* in_sizes, int n_in,
                              void* d_out, int out_size, void* d_ws, size_t ws_size,
                              hipStream_t stream) {
    (void)in_sizes; (void)n_in; (void)out_size; (void)ws_size;
    const float* x     = (const float*)d_in[0];
    const float* ggl_w = (const float*)d_in[1];
    const float* ggl_b = (const float*)d_in[2];
    const float* w1    = (const float*)d_in[3];
    const float* b1    = (const float*)d_in[4];
    const float* w2    = (const float*)d_in[5];
    const float* b2    = (const float*)d_in[6];
    const float* w3    = (const float*)d_in[7];
    const float* b3    = (const float*)d_in[8];
    const float* gamma = (const float*)d_in[9];
    const float* beta  = (const float*)d_in[10];

    float* out = (float*)d_out;
    float* x1 = out;
    float* x2 = out + (size_t)Nn * OUTC;
    float* x3 = out + 2 * (size_t)Nn * OUTC;
    float* An = out + 3 * (size_t)Nn * OUTC;

    float* ws = (float*)d_ws;
    float* attrM = ws;                       // attr, later reused for M  (N*N)
    size_t off = (size_t)Nn * Nn;
    float* tx1  = ws + off; off += (size_t)Nn * FEAT;
    float* tx2  = ws + off; off += (size_t)Nn * FEAT;
    float* rnrm = ws + off; off += Nn;
    float* dinv = ws + off; off += Nn;
    float* mn   = ws + off; off += OUTC;
    float* istd = ws + off; off += OUTC;

    dim3 blk(256);
    auto g2 = [](int m, int n) { return dim3((n + BN - 1) / BN, (m + BM - 1) / BM); };

    // 1) attr = sigmoid(x @ ggl_w + ggl_b)           [N,N]
    gemm_wmma_f32<<<g2(Nn, Nn), blk, 0, stream>>>(x, FEAT, ggl_w, Nn, 0, attrM, Nn,
        Nn, Nn, FEAT, 1.f, ggl_b, nullptr, 0, 0.f, nullptr, nullptr, 1, 0);
    // 2) rnrm[i] = 1/max(||attr_i||, eps)
    row_invnorm<<<Nn, blk, 0, stream>>>(attrM, Nn, rnrm);
    // 3) adj = (attr @ attr^T) * rnrm_i * rnrm_j  -> d_out A_norm slot (async fast path)
    gemm_wmma_f32<<<g2(Nn, Nn), blk, 0, stream>>>(attrM, Nn, attrM, Nn, 1, An, Nn,
        Nn, Nn, Nn, 1.f, nullptr, nullptr, 0, 0.f, rnrm, rnrm, 0, 0);
    // 4) row-max normalize in place; deg/dinv
    rowmax_norm_deg<<<Nn, blk, 0, stream>>>(An, Nn, dinv);
    // 5) M = -(dinv_i dinv_j A_norm[j,i]), zero diag   (reuses attr buffer)
    build_M<<<dim3(Nn / 32, Nn / 32), blk, 0, stream>>>(An, dinv, attrM, Nn);
    // 6) tx1 = M @ x
    gemm_wmma_f32<<<g2(Nn, FEAT), blk, 0, stream>>>(attrM, Nn, x, FEAT, 0, tx1, FEAT,
        Nn, FEAT, Nn, 1.f, nullptr, nullptr, 0, 0.f, nullptr, nullptr, 0, 0);
    // 7) tx2 = 2*(M @ tx1) - x
    gemm_wmma_f32<<<g2(Nn, FEAT), blk, 0, stream>>>(attrM, Nn, tx1, FEAT, 0, tx2, FEAT,
        Nn, FEAT, Nn, 2.f, nullptr, x, FEAT, -1.f, nullptr, nullptr, 0, 0);
    // 8) s1 = x@w1[0] + b1  -> x1
    gemm_wmma_f32<<<g2(Nn, OUTC), blk, 0, stream>>>(x, FEAT, w1, OUTC, 0, x1, OUTC,
        Nn, OUTC, FEAT, 1.f, b1, nullptr, 0, 0.f, nullptr, nullptr, 0, 0);
    // 9) s2 = x@w2[0] + b2 ; += tx1@w2[1]  -> x2
    gemm_wmma_f32<<<g2(Nn, OUTC), blk, 0, stream>>>(x, FEAT, w2, OUTC, 0, x2, OUTC,
        Nn, OUTC, FEAT, 1.f, b2, nullptr, 0, 0.f, nullptr, nullptr, 0, 0);
    gemm_wmma_f32<<<g2(Nn, OUTC), blk, 0, stream>>>(tx1, FEAT, w2 + (size_t)FEAT * OUTC, OUTC, 0, x2, OUTC,
        Nn, OUTC, FEAT, 1.f, nullptr, nullptr, 0, 0.f, nullptr, nullptr, 0, 1);
    // 10) s3 = x@w3[0] + b3 ; += tx1@w3[1] ; += tx2@w3[2]  -> x3
    gemm_wmma_f32<<<g2(Nn, OUTC), blk, 0, stream>>>(x, FEAT, w3, OUTC, 0, x3, OUTC,
        Nn, OUTC, FEAT, 1.f, b3, nullptr, 0, 0.f, nullptr, nullptr, 0, 0);
    gemm_wmma_f32<<<g2(Nn, OUTC), blk, 0, stream>>>(tx1, FEAT, w3 + (size_t)FEAT * OUTC, OUTC, 0, x3, OUTC,
        Nn, OUTC, FEAT, 1.f, nullptr, nullptr, 0, 0.f, nullptr, nullptr, 0, 1);
    gemm_wmma_f32<<<g2(Nn, OUTC), blk, 0, stream>>>(tx2, FEAT, w3 + 2 * (size_t)FEAT * OUTC, OUTC, 0, x3, OUTC,
        Nn, OUTC, FEAT, 1.f, nullptr, nullptr, 0, 0.f, nullptr, nullptr, 0, 1);

    // BN (training mode, biased var) in place on x1/x2/x3
    int nApply = (int)(((size_t)Nn * OUTC + 255) / 256);
    bn_stats<<<OUTC, blk, 0, stream>>>(x1, Nn, OUTC, mn, istd);
    bn_apply<<<nApply, blk, 0, stream>>>(x1, Nn, OUTC, mn, istd, gamma, beta);
    bn_stats<<<OUTC, blk, 0, stream>>>(x2, Nn, OUTC, mn, istd);
    bn_apply<<<nApply, blk, 0, stream>>>(x2, Nn, OUTC, mn, istd, gamma, beta);
    bn_stats<<<OUTC, blk, 0, stream>>>(x3, Nn, OUTC, mn, istd);
    bn_apply<<<nApply, blk, 0, stream>>>(x3, Nn, OUTC, mn, istd, gamma, beta);
}